// BiAttention_29540785062241
// MI455X (gfx1250) — compile-verified
//
#include <hip/hip_runtime.h>
#include <hip/hip_bf16.h>

typedef __attribute__((ext_vector_type(2))) float v2f;
typedef __attribute__((ext_vector_type(8))) float v8f;

#define BB 32
#define NN 1024
#define QQ 64
#define HH 768

// ---------------------------------------------------------------------------
// Kernel 0: column bias  cb[b,q] = qc[b,q,:] . w_q      (one wave per row)
// ---------------------------------------------------------------------------
__global__ __launch_bounds__(256) void k_colbias(const float* __restrict__ qc,
                                                 const float* __restrict__ w,
                                                 float* __restrict__ cb) {
    const int lane = threadIdx.x & 31;
    const int wid  = threadIdx.x >> 5;
    const int row  = blockIdx.x * 8 + wid;          // row in [0, B*Q)
    const float* src = qc + (size_t)row * HH;
    const float* wq  = w + HH;
    float s = 0.f;
    #pragma unroll 4
    for (int k = lane; k < HH; k += 32) s += src[k] * wq[k];
    #pragma unroll
    for (int off = 16; off > 0; off >>= 1) s += __shfl_xor(s, off, 32);
    if (lane == 0) cb[row] = s;
}

// ---------------------------------------------------------------------------
// Kernel 1: per 16-row tile of N (one wave per block):
//   sim = (nc * w_nq) @ qc^T + cb   via V_WMMA_F32_16X16X4_F32
//   softmax over q (in-register, shfl within 16-lane halves)
//   m[b,n] = rb[n] + max_q(...)    (rb accumulated in the K loop)
//   n2q = a @ qc                   via WMMA (a transposed through LDS)
//   writes out segments: [0:H]=nc, [H:2H]=n2q, [2H:3H]=nc*n2q
// ---------------------------------------------------------------------------
__global__ __launch_bounds__(32) void k_sim(const float* __restrict__ nc,
                                            const float* __restrict__ qc,
                                            const float* __restrict__ w,
                                            const float* __restrict__ cb,
                                            float* __restrict__ m_out,
                                            float* __restrict__ out) {
    __shared__ float aT[64 * 17];   // a transposed: aT[q*17 + row], pad avoids conflicts
    __shared__ float mstage[16];

    const int lane    = threadIdx.x;        // 0..31 (single wave)
    const int b       = blockIdx.y;
    const int rowBase = blockIdx.x * 16;
    const int lr      = lane & 15;          // A: row M ; B/C: column
    const int khi     = (lane >> 4) * 2;    // K sub-offset of this half

    const float* ncRow  = nc + (size_t)(b * NN + rowBase + lr) * HH;
    const float* qcBase = qc + (size_t)b * QQ * HH;
    const float* wnq    = w + 2 * HH;
    const float* wn     = w;

    v8f c0 = {}, c1 = {}, c2 = {}, c3 = {};
    float rb = 0.f;

    // ---- GEMM 1: sim = (nc*w_nq) @ qc^T , K = 768 in steps of 4 ----
    #pragma unroll 2
    for (int k = 0; k < HH; k += 4) {
        const int kk = k + khi;
        v2f av  = *(const v2f*)(ncRow + kk);
        v2f wv  = *(const v2f*)(wnq + kk);
        v2f wnv = *(const v2f*)(wn + kk);
        rb += av.x * wnv.x + av.y * wnv.y;
        v2f a; a.x = av.x * wv.x; a.y = av.y * wv.y;
        v2f b0 = *(const v2f*)(qcBase + (size_t)(0 * 16 + lr) * HH + kk);
        v2f b1 = *(const v2f*)(qcBase + (size_t)(1 * 16 + lr) * HH + kk);
        v2f b2 = *(const v2f*)(qcBase + (size_t)(2 * 16 + lr) * HH + kk);
        v2f b3 = *(const v2f*)(qcBase + (size_t)(3 * 16 + lr) * HH + kk);
        c0 = __builtin_amdgcn_wmma_f32_16x16x4_f32(false, a, false, b0, (short)0, c0, false, false);
        c1 = __builtin_amdgcn_wmma_f32_16x16x4_f32(false, a, false, b1, (short)0, c1, false, false);
        c2 = __builtin_amdgcn_wmma_f32_16x16x4_f32(false, a, false, b2, (short)0, c2, false, false);
        c3 = __builtin_amdgcn_wmma_f32_16x16x4_f32(false, a, false, b3, (short)0, c3, false, false);
    }

    // ---- add column bias (lane holds one q column per tile) ----
    const float cb0 = cb[b * QQ + 0 * 16 + lr];
    const float cb1 = cb[b * QQ + 1 * 16 + lr];
    const float cb2 = cb[b * QQ + 2 * 16 + lr];
    const float cb3 = cb[b * QQ + 3 * 16 + lr];
    #pragma unroll
    for (int r = 0; r < 8; ++r) { c0[r] += cb0; c1[r] += cb1; c2[r] += cb2; c3[r] += cb3; }

    // ---- row max over q (reduce across 16 lanes of each half) ----
    float mx[8];
    #pragma unroll
    for (int r = 0; r < 8; ++r)
        mx[r] = fmaxf(fmaxf(c0[r], c1[r]), fmaxf(c2[r], c3[r]));
    #pragma unroll
    for (int off = 1; off < 16; off <<= 1) {
        #pragma unroll
        for (int r = 0; r < 8; ++r) mx[r] = fmaxf(mx[r], __shfl_xor(mx[r], off, 32));
    }

    // ---- exp + row sum ----
    float sum[8];
    #pragma unroll
    for (int r = 0; r < 8; ++r) {
        c0[r] = __expf(c0[r] - mx[r]);
        c1[r] = __expf(c1[r] - mx[r]);
        c2[r] = __expf(c2[r] - mx[r]);
        c3[r] = __expf(c3[r] - mx[r]);
        sum[r] = c0[r] + c1[r] + c2[r] + c3[r];
    }
    #pragma unroll
    for (int off = 1; off < 16; off <<= 1) {
        #pragma unroll
        for (int r = 0; r < 8; ++r) sum[r] += __shfl_xor(sum[r], off, 32);
    }

    // ---- m[b,n] = rowmax + rb ----
    float rbF = rb + __shfl_xor(rb, 16, 32);     // lane L (<16) now has full rb of row L
    if (lane == 0) {
        #pragma unroll
        for (int r = 0; r < 8; ++r) mstage[r] = mx[r];
    }
    if (lane == 16) {
        #pragma unroll
        for (int r = 0; r < 8; ++r) mstage[8 + r] = mx[r];
    }
    __syncthreads();
    if (lane < 16) m_out[b * NN + rowBase + lane] = mstage[lane] + rbF;

    // ---- store normalized a, transposed, to LDS:  aT[q*17 + row] ----
    const int mb = (lane >> 4) * 8;
    #pragma unroll
    for (int r = 0; r < 8; ++r) {
        float inv = 1.0f / sum[r];
        aT[(0 * 16 + lr) * 17 + mb + r] = c0[r] * inv;
        aT[(1 * 16 + lr) * 17 + mb + r] = c1[r] * inv;
        aT[(2 * 16 + lr) * 17 + mb + r] = c2[r] * inv;
        aT[(3 * 16 + lr) * 17 + mb + r] = c3[r] * inv;
    }
    __syncthreads();

    // ---- GEMM 2: n2q(16x768) = a(16x64) @ qc(64x768), fused epilogue ----
    const size_t obase = (size_t)(b * NN + rowBase) * (4 * HH);
    for (int ct = 0; ct < HH / 16; ++ct) {
        const int h0 = ct * 16;
        v8f acc = {};
        #pragma unroll
        for (int kk = 0; kk < QQ; kk += 4) {
            const int k = kk + khi;
            v2f a, bv;
            a.x  = aT[(k + 0) * 17 + lr];
            a.y  = aT[(k + 1) * 17 + lr];
            bv.x = qcBase[(size_t)(k + 0) * HH + h0 + lr];
            bv.y = qcBase[(size_t)(k + 1) * HH + h0 + lr];
            acc = __builtin_amdgcn_wmma_f32_16x16x4_f32(false, a, false, bv, (short)0, acc, false, false);
        }
        const int h = h0 + lr;
        #pragma unroll
        for (int r = 0; r < 8; ++r) {
            const int row = mb + r;
            const size_t orow = obase + (size_t)row * (4 * HH);
            const float ncv = nc[(size_t)(b * NN + rowBase + row) * HH + h];
            const float nq  = acc[r];
            out[orow + 0 * HH + h] = ncv;
            out[orow + 1 * HH + h] = nq;
            out[orow + 2 * HH + h] = ncv * nq;
        }
    }
}

// ---------------------------------------------------------------------------
// Kernel 2: per batch: bw = softmax_n(m);  q2n[b,h] = sum_n bw[n]*nc[b,n,h]
// ---------------------------------------------------------------------------
__global__ __launch_bounds__(256) void k_q2n(const float* __restrict__ nc,
                                             const float* __restrict__ m,
                                             float* __restrict__ q2n) {
    __shared__ float wts[NN];
    __shared__ float red[256];
    const int b = blockIdx.x, tid = threadIdx.x;
    const float* mb = m + b * NN;

    float v0 = mb[tid], v1 = mb[tid + 256], v2 = mb[tid + 512], v3 = mb[tid + 768];
    float mx = fmaxf(fmaxf(v0, v1), fmaxf(v2, v3));
    red[tid] = mx; __syncthreads();
    #pragma unroll
    for (int s = 128; s > 0; s >>= 1) {
        if (tid < s) red[tid] = fmaxf(red[tid], red[tid + s]);
        __syncthreads();
    }
    mx = red[0]; __syncthreads();

    float e0 = __expf(v0 - mx), e1 = __expf(v1 - mx), e2 = __expf(v2 - mx), e3 = __expf(v3 - mx);
    wts[tid] = e0; wts[tid + 256] = e1; wts[tid + 512] = e2; wts[tid + 768] = e3;
    red[tid] = e0 + e1 + e2 + e3; __syncthreads();
    #pragma unroll
    for (int s = 128; s > 0; s >>= 1) {
        if (tid < s) red[tid] += red[tid + s];
        __syncthreads();
    }
    const float inv = 1.0f / red[0];

    const float* ncb = nc + (size_t)b * NN * HH;
    float a0 = 0.f, a1 = 0.f, a2 = 0.f;
    for (int n = 0; n < NN; ++n) {
        const float wv = wts[n];
        const float* r = ncb + (size_t)n * HH;
        a0 = fmaf(wv, r[tid], a0);
        a1 = fmaf(wv, r[tid + 256], a1);
        a2 = fmaf(wv, r[tid + 512], a2);
    }
    q2n[b * HH + tid]       = a0 * inv;
    q2n[b * HH + tid + 256] = a1 * inv;
    q2n[b * HH + tid + 512] = a2 * inv;
}

// ---------------------------------------------------------------------------
// Kernel 3: out[b,n,3H+h] = nc[b,n,h] * q2n[b,h]    (float4 elementwise)
// ---------------------------------------------------------------------------
__global__ __launch_bounds__(256) void k_seg3(const float* __restrict__ nc,
                                              const float* __restrict__ q2n,
                                              float* __restrict__ out) {
    const size_t idx = (size_t)blockIdx.x * 256 + threadIdx.x;   // quads
    const size_t e   = idx * 4;                                  // element index
    const int    h   = (int)(e % HH);
    const size_t row = e / HH;                                   // b*N + n
    const int    b   = (int)(row / NN);
    const float4 a = *(const float4*)(nc + e);
    const float4 q = *(const float4*)(q2n + (size_t)b * HH + h);
    float4 r;
    r.x = a.x * q.x; r.y = a.y * q.y; r.z = a.z * q.z; r.w = a.w * q.w;
    *(float4*)(out + row * (4 * HH) + 3 * HH + h) = r;
}

// ---------------------------------------------------------------------------
extern "C" void kernel_launch(void* const* d_in, const int* in_sizes, int n_in,
                              void* d_out, int out_size, void* d_ws, size_t ws_size,
                              hipStream_t stream) {
    const float* nc = (const float*)d_in[0];   // (B,N,H)
    const float* qc = (const float*)d_in[1];   // (B,Q,H)
    // d_in[2] = nodes_hidden: unused by the reference
    const float* w  = (const float*)d_in[3];   // (3H,)
    float* out = (float*)d_out;

    float* ws  = (float*)d_ws;
    float* cb  = ws;                       // B*Q   = 2048
    float* m   = ws + BB * QQ;             // B*N   = 32768
    float* q2n = m + BB * NN;              // B*H   = 24576

    // cb[b,q]
    k_colbias<<<dim3((BB * QQ) / 8), dim3(256), 0, stream>>>(qc, w, cb);
    // sim + softmax_q + n2q + out segments 0..2, and m[b,n]
    k_sim<<<dim3(NN / 16, BB), dim3(32), 0, stream>>>(nc, qc, w, cb, m, out);
    // softmax_n(m) -> q2n[b,h]
    k_q2n<<<dim3(BB), dim3(256), 0, stream>>>(nc, m, q2n);
    // out segment 3
    k_seg3<<<dim3((BB * NN * HH / 4) / 256), dim3(256), 0, stream>>>(nc, q2n, out);
}